// DechirpSTFT_28132035788956
// MI455X (gfx1250) — compile-verified
//
#include <hip/hip_runtime.h>
#include <math.h>

// ---------------------------------------------------------------------------
// Dechirped STFT, FFT-1024 as WMMA fp32 GEMMs (4-step, 1024 = 16 x 64)
// Target: gfx1250 (CDNA5), wave32, V_WMMA_F32_16X16X4_F32
//   Stage 1: Y(64x16) = F64(64x64) x C(64x16)      -> 256 WMMA / frame
//   Twiddle: T[n1][k2] = Y[k2][n1] * W1024^{n1 k2}  (register + LDS transpose)
//   Stage 2: X(16x64) = F16(16x16) x T(16x64)      -> 64 WMMA / frame
// ---------------------------------------------------------------------------

typedef float v2f __attribute__((ext_vector_type(2)));
typedef float v8f __attribute__((ext_vector_type(8)));

#define K_FFT  1024
#define HOPSZ  512
#define NBATCH 2
#define NWIN   1023
#define NFREQ  16
#define TOTAL_FRAMES (NFREQ * NBATCH * NWIN)   // 32736
#define WPB    4                               // wave32s per block (1 frame/wave)
#define PI_F   3.14159265358979323846f
#define CHIRP_A 0.5f

// workspace layout (byte offsets into d_ws)
#define OFF_IDX    0                          // int   [16*1024]  resample idx_lo
#define OFF_WTAB   (64*1024)                  // f4    [16*1024]  {wlo*chre, wlo*chim, whi*chre, whi*chim}
#define OFF_F64P4  (OFF_WTAB  + 256*1024)     // f4    [64*32]    {re0,re1,im0,im1} per (row,kpair)
#define OFF_F64P2  (OFF_F64P4 + 32*1024)      // f2    [64*32]    {-im0,-im1}
#define OFF_F16P4  (OFF_F64P2 + 16*1024)      // f4    [16*8]
#define OFF_F16P2  (OFF_F16P4 + 2*1024)       // f2    [16*8]
#define OFF_TW     (OFF_F16P2 + 1*1024)       // f2    [16*64]    twiddle {re,im} (n1*64+k2)

// ---------------------------------------------------------------------------
// Setup: resample weights (Hann + chirp folded in) and packed DFT/twiddle
// tables in WMMA A-fragment-friendly order.
// ---------------------------------------------------------------------------
__global__ void dechirp_setup_kernel(const float* __restrict__ dlnf,
                                     char* __restrict__ ws)
{
    const int tid = blockIdx.x * blockDim.x + threadIdx.x;

    int*    idx_tab = (int*)   (ws + OFF_IDX);
    float4* wtab    = (float4*)(ws + OFF_WTAB);
    float4* f64p4   = (float4*)(ws + OFF_F64P4);
    float2* f64p2   = (float2*)(ws + OFF_F64P2);
    float4* f16p4   = (float4*)(ws + OFF_F16P4);
    float2* f16p2   = (float2*)(ws + OFF_F16P2);
    float2* twp     = (float2*)(ws + OFF_TW);

    if (tid < NFREQ * K_FFT) {
        const int f = tid >> 10;
        const int k = tid & (K_FFT - 1);

        const float beta = 2.0f * dlnf[f];
        const bool  safe = fabsf(beta) < 1e-8f;
        const float bs   = safe ? 1e-8f : beta;
        const float tau  = (float)k * (1.0f / 1023.0f);
        const float em1  = expf(bs) - 1.0f;
        const float ident = -1.0f + 2.0f * (float)k * (1.0f / 1023.0f);
        float t_src = (2.0f / bs) * log1pf(tau * em1) - 1.0f;
        if (safe) t_src = ident;

        const float idxf = (t_src + 1.0f) * 0.5f * 1023.0f;
        int il = (int)idxf;
        il = il < 0 ? 0 : (il > 1022 ? 1022 : il);
        const float frac = idxf - (float)il;

        const float h0 = 0.5f * (1.0f - cosf(2.0f * PI_F * (float)il       * (1.0f / 1024.0f)));
        const float h1 = 0.5f * (1.0f - cosf(2.0f * PI_F * (float)(il + 1) * (1.0f / 1024.0f)));
        const float wlo = h0 * (1.0f - frac);
        const float whi = h1 * frac;

        // chirp at destination sample k: exp(-i * A * t^2)
        const float ang  = CHIRP_A * ident * ident;
        const float chre = cosf(ang);
        const float chim = -sinf(ang);

        idx_tab[tid] = il;
        wtab[tid] = make_float4(wlo * chre, wlo * chim, whi * chre, whi * chim);
    }

    if (tid < 64 * 32) {            // DFT64[row][k] = exp(-2πi row k / 64), packed per k-pair
        const int row = tid >> 5, kp = tid & 31;
        const int k0 = kp * 2, k1 = k0 + 1;
        const float a0 = -2.0f * PI_F * (float)((row * k0) & 63) * (1.0f / 64.0f);
        const float a1 = -2.0f * PI_F * (float)((row * k1) & 63) * (1.0f / 64.0f);
        const float c0 = cosf(a0), s0 = sinf(a0);
        const float c1 = cosf(a1), s1 = sinf(a1);
        f64p4[tid] = make_float4(c0, c1, s0, s1);
        f64p2[tid] = make_float2(-s0, -s1);
    }
    if (tid < 16 * 8) {             // DFT16[row][k] = exp(-2πi row k / 16), packed per k-pair
        const int row = tid >> 3, kp = tid & 7;
        const int k0 = kp * 2, k1 = k0 + 1;
        const float a0 = -2.0f * PI_F * (float)((row * k0) & 15) * (1.0f / 16.0f);
        const float a1 = -2.0f * PI_F * (float)((row * k1) & 15) * (1.0f / 16.0f);
        const float c0 = cosf(a0), s0 = sinf(a0);
        const float c1 = cosf(a1), s1 = sinf(a1);
        f16p4[tid] = make_float4(c0, c1, s0, s1);
        f16p2[tid] = make_float2(-s0, -s1);
    }
    if (tid < 16 * 64) {            // tw[n1][k2] = exp(-2πi n1 k2 / 1024), interleaved
        const int n1 = tid >> 6, k2 = tid & 63;
        const float ang = -2.0f * PI_F * (float)((n1 * k2) & 1023) * (1.0f / 1024.0f);
        twp[tid] = make_float2(cosf(ang), sinf(ang));
    }
}

// ---------------------------------------------------------------------------
// Main kernel: one wave32 per frame. Resample+chirp front end -> LDS planes,
// FFT1024 = F16 * (TW .* (F64 * C)) as fp32 WMMA GEMMs.
// ---------------------------------------------------------------------------
__global__ __launch_bounds__(WPB * 32)
void dechirp_stft_wmma_kernel(const float* __restrict__ x,
                              const char*  __restrict__ ws,
                              float* __restrict__ out)
{
    // per-wave frame buffer, de-interleaved: re plane [0..1023], im plane [1024..2047]
    __shared__ float lds[WPB][2 * K_FFT];

    const int lane = threadIdx.x & 31;
    const int wv   = threadIdx.x >> 5;

    // frame decode: F = f*(2*1023) + b*1023 + w ; grid covers exactly 32736
    const int F   = blockIdx.x * WPB + wv;
    const int f   = F / (NBATCH * NWIN);
    const int rem = F - f * (NBATCH * NWIN);
    const int b   = rem / NWIN;
    const int wi  = rem - b * NWIN;

    const int*    idx_tab = (const int*)   (ws + OFF_IDX)  + f * K_FFT;
    const float4* wtab    = (const float4*)(ws + OFF_WTAB) + f * K_FFT;
    const float4* f64p4   = (const float4*)(ws + OFF_F64P4);
    const float2* f64p2   = (const float2*)(ws + OFF_F64P2);
    const float4* f16p4   = (const float4*)(ws + OFF_F16P4);
    const float2* f16p2   = (const float2*)(ws + OFF_F16P2);
    const float2* twp     = (const float2*)(ws + OFF_TW);

    const float* xrow = x + (size_t)b * 524288 + (size_t)wi * HOPSZ;
    float* rebuf = &lds[wv][0];
    float* imbuf = &lds[wv][K_FFT];

    // ---- front end: gather, resample, window+chirp (folded weights) ----
    for (int k = lane; k < K_FFT; k += 32) {
        const int    il = idx_tab[k];
        const float4 w4 = wtab[k];
        const float xlo = xrow[il];
        const float xhi = xrow[il + 1];
        rebuf[k] = xlo * w4.x + xhi * w4.z;
        imbuf[k] = xlo * w4.y + xhi * w4.w;
    }

    const int lrow  = lane & 15;              // fragment row/col index
    const int half  = lane >> 4;              // 0: K0/K1 half, 1: K2/K3 half
    const int rbase = half << 3;              // C/D rows r / r+8

    // ---- Stage 1: Y(64x16) = F64(64x64) * C(64x16), C[n2][n1] = c[n1+16*n2]
    v8f accRe[4], accIm[4];
    #pragma unroll
    for (int m = 0; m < 4; ++m) { accRe[m] = (v8f)0.0f; accIm[m] = (v8f)0.0f; }

    for (int s = 0; s < 16; ++s) {
        const int kb  = s * 4 + half * 2;
        const int kp  = s * 2 + half;         // packed k-pair index
        const int off = kb * 16 + lrow;
        const v2f bre = {rebuf[off], rebuf[off + 16]};   // -> ds_load_2addr_b32
        const v2f bim = {imbuf[off], imbuf[off + 16]};
        #pragma unroll
        for (int m = 0; m < 4; ++m) {
            const int idx = (m * 16 + lrow) * 32 + kp;
            const float4 a4 = f64p4[idx];
            const float2 an = f64p2[idx];
            const v2f are  = {a4.x, a4.y};
            const v2f aim  = {a4.z, a4.w};
            const v2f aimn = {an.x, an.y};
            accRe[m] = __builtin_amdgcn_wmma_f32_16x16x4_f32(false, are,  false, bre, (short)0, accRe[m], false, false);
            accRe[m] = __builtin_amdgcn_wmma_f32_16x16x4_f32(false, aimn, false, bim, (short)0, accRe[m], false, false);
            accIm[m] = __builtin_amdgcn_wmma_f32_16x16x4_f32(false, are,  false, bim, (short)0, accIm[m], false, false);
            accIm[m] = __builtin_amdgcn_wmma_f32_16x16x4_f32(false, aim,  false, bre, (short)0, accIm[m], false, false);
        }
    }

    // ---- twiddle + transpose through LDS: T[n1][k2] = Y[k2][n1] * tw ----
    // destination elements per tile are contiguous: base..base+7 in each plane
    #pragma unroll
    for (int m = 0; m < 4; ++m) {
        const int base = lrow * 64 + m * 16 + rbase;     // multiple of 8
        const float4* t4 = (const float4*)(twp + base);  // 4 x b128, coalesced
        float tre[8], tim[8];
        #pragma unroll
        for (int j = 0; j < 4; ++j) {
            const float4 t = t4[j];
            const float y0r = accRe[m][2*j],     y0i = accIm[m][2*j];
            const float y1r = accRe[m][2*j + 1], y1i = accIm[m][2*j + 1];
            tre[2*j]     = y0r * t.x - y0i * t.y;
            tim[2*j]     = y0r * t.y + y0i * t.x;
            tre[2*j + 1] = y1r * t.z - y1i * t.w;
            tim[2*j + 1] = y1r * t.w + y1i * t.z;
        }
        ((float4*)(rebuf + base))[0] = make_float4(tre[0], tre[1], tre[2], tre[3]);
        ((float4*)(rebuf + base))[1] = make_float4(tre[4], tre[5], tre[6], tre[7]);
        ((float4*)(imbuf + base))[0] = make_float4(tim[0], tim[1], tim[2], tim[3]);
        ((float4*)(imbuf + base))[1] = make_float4(tim[4], tim[5], tim[6], tim[7]);
    }

    // ---- Stage 2: X(16x64) = F16(16x16) * T(16x64); X[64*k1 + k2] ----
    const size_t outBase = ((size_t)((f * NBATCH + b) * NWIN + wi)) * K_FFT;
    float2* outc = (float2*)out + outBase;

    for (int nt = 0; nt < 4; ++nt) {
        v8f dre = (v8f)0.0f, dim_ = (v8f)0.0f;
        const int k2g = nt * 16 + lrow;
        #pragma unroll
        for (int s = 0; s < 4; ++s) {
            const int kb  = s * 4 + half * 2;
            const int off = kb * 64 + k2g;
            const v2f bre = {rebuf[off], rebuf[off + 64]};
            const v2f bim = {imbuf[off], imbuf[off + 64]};
            const int idx = lrow * 8 + s * 2 + half;
            const float4 a4 = f16p4[idx];
            const float2 an = f16p2[idx];
            const v2f are  = {a4.x, a4.y};
            const v2f aim  = {a4.z, a4.w};
            const v2f aimn = {an.x, an.y};
            dre  = __builtin_amdgcn_wmma_f32_16x16x4_f32(false, are,  false, bre, (short)0, dre,  false, false);
            dre  = __builtin_amdgcn_wmma_f32_16x16x4_f32(false, aimn, false, bim, (short)0, dre,  false, false);
            dim_ = __builtin_amdgcn_wmma_f32_16x16x4_f32(false, are,  false, bim, (short)0, dim_, false, false);
            dim_ = __builtin_amdgcn_wmma_f32_16x16x4_f32(false, aim,  false, bre, (short)0, dim_, false, false);
        }
        // streamed output: written exactly once -> non-temporal stores
        #pragma unroll
        for (int r = 0; r < 8; ++r) {
            const int k1 = r + rbase;
            const v2f val = {dre[r], dim_[r]};
            __builtin_nontemporal_store(val, (v2f*)(outc + (size_t)k1 * 64 + k2g));
        }
    }
}

// ---------------------------------------------------------------------------
extern "C" void kernel_launch(void* const* d_in, const int* in_sizes, int n_in,
                              void* d_out, int out_size, void* d_ws, size_t ws_size,
                              hipStream_t stream)
{
    (void)in_sizes; (void)n_in; (void)out_size; (void)ws_size;
    const float* x    = (const float*)d_in[0];
    const float* dlnf = (const float*)d_in[1];
    float* out = (float*)d_out;
    char*  ws  = (char*)d_ws;

    dechirp_setup_kernel<<<dim3(64), dim3(256), 0, stream>>>(dlnf, ws);

    dechirp_stft_wmma_kernel<<<dim3(TOTAL_FRAMES / WPB), dim3(WPB * 32), 0, stream>>>(x, ws, out);
}